// Process_24979529793585
// MI455X (gfx1250) — compile-verified
//
#include <hip/hip_runtime.h>
#include <math.h>

typedef __attribute__((ext_vector_type(2))) float v2f;
typedef __attribute__((ext_vector_type(8))) float v8f;

#define B_     32
#define H_     1024
#define L_     1024
#define N4H    4096
#define STEPS_ 10

// ---------------------------------------------------------------------------
// Msum[b,h] = sum_l M[b,h,l]  (the degenerate attention read; also output r)
// One wave32 per 1024-float row; coalesced 128B strided loads + shuffle tree.
// ---------------------------------------------------------------------------
__global__ __launch_bounds__(256) void rowsum_kernel(const float* __restrict__ M,
                                                     float* __restrict__ Msum) {
  const int row  = (blockIdx.x * blockDim.x + threadIdx.x) >> 5;
  const int lane = threadIdx.x & 31;
  if (row >= B_ * H_) return;
  const float* r = M + (size_t)row * L_;
  float s = 0.f;
#pragma unroll 4
  for (int i = lane; i < L_; i += 32) s += r[i];
#pragma unroll
  for (int off = 16; off > 0; off >>= 1) s += __shfl_down(s, off, 32);
  if (lane == 0) Msum[row] = s;
}

// ---------------------------------------------------------------------------
// Broadcast h0/c0 across the batch.
// ---------------------------------------------------------------------------
__global__ void init_hc(const float* __restrict__ h0, const float* __restrict__ c0,
                        float* __restrict__ h, float* __restrict__ c) {
  int idx = blockIdx.x * blockDim.x + threadIdx.x;
  if (idx >= B_ * H_) return;
  int j = idx & (H_ - 1);
  h[idx] = h0[j];
  c[idx] = c0[j];
}

// ---------------------------------------------------------------------------
// out[b,n] = sum_k X[b,k]*W[n,k] + base
//   base = Abase ? Abase[b,n] : bias1[n]+bias2[n]
// M=32, N=4096, K=1024. One wave per 16x16 tile via V_WMMA_F32_16X16X4_F32.
// Fragment layout (ISA 7.12.2):
//   A 16x4 : lane&15 = M row, (lane>>4)*2+v = K      -> contiguous float2/lane
//   B 4x16 : lane&15 = N col, (lane>>4)*2+v = K      -> contiguous float2/lane
//   D      : lane<16 -> (M=v,   N=lane), lane>=16 -> (M=v+8, N=lane-16)
// Two accumulators interleave independent WMMA chains.
// ---------------------------------------------------------------------------
__global__ __launch_bounds__(256) void gemm_gates(const float* __restrict__ X,
                                                  const float* __restrict__ W,
                                                  const float* __restrict__ Abase,
                                                  const float* __restrict__ bias1,
                                                  const float* __restrict__ bias2,
                                                  float* __restrict__ out) {
  const int wave   = (blockIdx.x * blockDim.x + threadIdx.x) >> 5;
  const int lane   = threadIdx.x & 31;
  const int mt     = wave >> 8;     // 0..1   (batch tile)
  const int nt     = wave & 255;    // 0..255 (gate-column tile)
  const int b0     = mt << 4;
  const int n0     = nt << 4;
  const int lane16 = lane & 15;
  const int hi     = lane >> 4;
  const int koff   = hi << 1;

  const float* xrow = X + (size_t)(b0 + lane16) * H_ + koff;
  const float* wrow = W + (size_t)(n0 + lane16) * H_ + koff;

  v8f acc0 = {};
  v8f acc1 = {};
#pragma unroll 8
  for (int k0 = 0; k0 < H_; k0 += 8) {
    v2f a0 = *(const v2f*)(xrow + k0);
    v2f w0 = *(const v2f*)(wrow + k0);
    v2f a1 = *(const v2f*)(xrow + k0 + 4);
    v2f w1 = *(const v2f*)(wrow + k0 + 4);
    acc0 = __builtin_amdgcn_wmma_f32_16x16x4_f32(false, a0, false, w0,
                                                 (short)0, acc0, false, false);
    acc1 = __builtin_amdgcn_wmma_f32_16x16x4_f32(false, a1, false, w1,
                                                 (short)0, acc1, false, false);
  }

  const int ncol = n0 + lane16;
#pragma unroll
  for (int v = 0; v < 8; ++v) {
    const int    mrow = b0 + v + (hi << 3);
    const size_t o    = (size_t)mrow * N4H + ncol;
    const float  base = Abase ? Abase[o] : (bias1[ncol] + bias2[ncol]);
    out[o] = acc0[v] + acc1[v] + base;
  }
}

// ---------------------------------------------------------------------------
// LSTM cell pointwise update (gate order i,f,g,o like torch.LSTMCell).
// ---------------------------------------------------------------------------
__global__ void lstm_cell(const float* __restrict__ gates,
                          float* __restrict__ h, float* __restrict__ c) {
  int idx = blockIdx.x * blockDim.x + threadIdx.x;
  if (idx >= B_ * H_) return;
  int b = idx >> 10;
  int j = idx & (H_ - 1);
  const float* g = gates + (size_t)b * N4H;
  float gi = g[j], gf = g[H_ + j], gg = g[2 * H_ + j], go = g[3 * H_ + j];
  float si = 1.f / (1.f + expf(-gi));
  float sf = 1.f / (1.f + expf(-gf));
  float so = 1.f / (1.f + expf(-go));
  float cn = sf * c[idx] + si * tanhf(gg);
  float hn = so * tanhf(cn);
  c[idx] = cn;
  h[idx] = hn;
}

// ---------------------------------------------------------------------------
// Pack outputs: out = concat(r = Msum, c_final), each B_*H_ floats.
// ---------------------------------------------------------------------------
__global__ void writeout(const float* __restrict__ Msum, const float* __restrict__ c,
                         float* __restrict__ out) {
  int idx = blockIdx.x * blockDim.x + threadIdx.x;
  if (idx < B_ * H_)
    out[idx] = Msum[idx];
  else if (idx < 2 * B_ * H_)
    out[idx] = c[idx - B_ * H_];
}

extern "C" void kernel_launch(void* const* d_in, const int* in_sizes, int n_in,
                              void* d_out, int out_size, void* d_ws, size_t ws_size,
                              hipStream_t stream) {
  const float* M   = (const float*)d_in[0];
  const float* Wih = (const float*)d_in[1];
  const float* Whh = (const float*)d_in[2];
  const float* bih = (const float*)d_in[3];
  const float* bhh = (const float*)d_in[4];
  const float* h0  = (const float*)d_in[5];
  const float* c0  = (const float*)d_in[6];
  float* out = (float*)d_out;

  float* ws    = (float*)d_ws;
  float* Msum  = ws;                 // B_*H_
  float* A     = Msum + B_ * H_;     // B_*N4H  (Msum@Wih^T + b_ih + b_hh)
  float* gates = A + B_ * N4H;       // B_*N4H
  float* h     = gates + B_ * N4H;   // B_*H_
  float* c     = h + B_ * H_;        // B_*H_

  // 1) degenerate attention: r == row-sum of M (constant across steps)
  rowsum_kernel<<<(B_ * H_) / 8, 256, 0, stream>>>(M, Msum);
  init_hc<<<(B_ * H_) / 256, 256, 0, stream>>>(h0, c0, h, c);

  // 2) step-invariant input-gate term: A = Msum @ Wih^T + (b_ih + b_hh)
  gemm_gates<<<64, 256, 0, stream>>>(Msum, Wih, nullptr, bih, bhh, A);

  // 3) 10 recurrent steps; step 0 has r == 0, so only the bias base applies.
  for (int t = 0; t < STEPS_; ++t) {
    const float* base = (t == 0) ? nullptr : A;
    gemm_gates<<<64, 256, 0, stream>>>(h, Whh, base, bih, bhh, gates);
    lstm_cell<<<(B_ * H_) / 256, 256, 0, stream>>>(gates, h, c);
  }

  // 4) outputs (r, c)
  writeout<<<(2 * B_ * H_) / 256, 256, 0, stream>>>(Msum, c, out);
}